// MtanGruRegr_90460601188985
// MI455X (gfx1250) — compile-verified
//
#include <hip/hip_runtime.h>
#include <math.h>

// ---------------------------------------------------------------------------
// mTAN attention + GRU + MLP head for MI455X (gfx1250), fp32 WMMA path.
// All matrix work uses V_WMMA_F32_16X16X4_F32 (wave32, 16x16 C/D tiles).
// ---------------------------------------------------------------------------

typedef float v2f __attribute__((ext_vector_type(2)));
typedef float v8f __attribute__((ext_vector_type(8)));

#define B_  32
#define LK_ 2048
#define D_  32
#define LQ_ 128
#define ET_ 128
#define H_  8
#define ETK_ 16
#define NH_ 128

__device__ __forceinline__ v8f wmma_f32(v2f a, v2f b, v8f c) {
  // D = A(16x4) x B(4x16) + C(16x16), fp32
  return __builtin_amdgcn_wmma_f32_16x16x4_f32(
      /*neg_a=*/false, a, /*neg_b=*/false, b,
      /*c_mod=*/(short)0, c, /*reuse_a=*/false, /*reuse_b=*/false);
}

// ---------------------------------------------------------------------------
// Fused time-embedding + projection:  C[m][n] = sum_e embed(t[m],e)*W[n][e]+b[n]
// embed(t,0)=t*Wlin+blin ; embed(t,e)=sin(t*Wper[e-1]+bper[e-1]) for e>=1.
// |t*Wper+bper| < ~0.3 here, so the hardware v_sin_f32 (__sinf) is accurate —
// avoids libm's Payne-Hanek reduction path entirely.
// K = N = ET = 128.  One wave per 16x16 output tile.
// ---------------------------------------------------------------------------
__global__ void proj_embed_gemm(const float* __restrict__ tvec,
                                const float* __restrict__ Wlin,
                                const float* __restrict__ blin,
                                const float* __restrict__ Wper,
                                const float* __restrict__ bper,
                                const float* __restrict__ W,
                                const float* __restrict__ bias,
                                float* __restrict__ C, int M) {
  const int mt = blockIdx.x, nt = blockIdx.y;
  const int lane = threadIdx.x, lm = lane & 15, hlf = lane >> 4;
  const int row = mt * 16 + lm;          // A row (M dim)
  const int col = nt * 16 + lm;          // B col (N dim)
  const float t  = tvec[row];
  const float wl = Wlin[0], bl = blin[0];

  v8f acc = {};
  for (int g = 0; g < ET_ / 4; ++g) {
    v2f a, b;
#pragma unroll
    for (int j = 0; j < 2; ++j) {
      const int e = g * 4 + hlf * 2 + j;
      a[j] = (e == 0) ? (t * wl + bl) : __sinf(t * Wper[e - 1] + bper[e - 1]);
      b[j] = W[(size_t)col * ET_ + e];
    }
    acc = wmma_f32(a, b, acc);
  }
#pragma unroll
  for (int j = 0; j < 8; ++j) {
    const int m = mt * 16 + j + 8 * hlf;
    const int n = nt * 16 + lm;
    C[(size_t)m * ET_ + n] = acc[j] + bias[n];
  }
}

// ---------------------------------------------------------------------------
// Generic masked WMMA GEMM:  C[m][n] = relu?( sum_k A[m][k]*W[n][k] + bias[n] )
// A: MxK row-major, W: NxK row-major, C: MxN row-major. Handles ragged tiles.
// ---------------------------------------------------------------------------
__global__ void gemm_wmma(const float* __restrict__ A,
                          const float* __restrict__ W,
                          const float* __restrict__ bias,
                          float* __restrict__ C,
                          int M, int N, int K, int relu) {
  const int mt = blockIdx.x, nt = blockIdx.y;
  const int lane = threadIdx.x, lm = lane & 15, hlf = lane >> 4;
  const int row = mt * 16 + lm;
  const int col = nt * 16 + lm;
  const int KG = (K + 3) >> 2;

  v8f acc = {};
  for (int g = 0; g < KG; ++g) {
    v2f a, b;
#pragma unroll
    for (int j = 0; j < 2; ++j) {
      const int kk = g * 4 + hlf * 2 + j;
      a[j] = (row < M && kk < K) ? A[(size_t)row * K + kk] : 0.0f;
      b[j] = (col < N && kk < K) ? W[(size_t)col * K + kk] : 0.0f;
    }
    acc = wmma_f32(a, b, acc);
  }
#pragma unroll
  for (int j = 0; j < 8; ++j) {
    const int m = mt * 16 + j + 8 * hlf;
    const int n = nt * 16 + lm;
    if (m < M && n < N) {
      float v = acc[j] + bias[n];
      if (relu) v = fmaxf(v, 0.0f);
      C[(size_t)m * N + n] = v;
    }
  }
}

// ---------------------------------------------------------------------------
// Flash-style attention: one wave per (q-tile of 16, head, batch).
// scores = (qh . kh^T)/4  -> online softmax over LK -> acc += P @ x.
// Never materializes (B,H,LQ,LK). Output att layout: (B, LQ, H*D).
// ---------------------------------------------------------------------------
__global__ void attention_kernel(const float* __restrict__ qh,  // (LQ, ET)
                                 const float* __restrict__ kh,  // (B*LK, ET)
                                 const float* __restrict__ x,   // (B, LK, D)
                                 float* __restrict__ att) {     // (B, LQ, H*D)
  const int qt = blockIdx.x;   // 0..LQ/16-1
  const int h  = blockIdx.y;   // 0..H-1
  const int b  = blockIdx.z;   // 0..B-1
  const int lane = threadIdx.x, lm = lane & 15, hlf = lane >> 4;

  __shared__ float sS[16 * 16];
  __shared__ float scorr[16];
  __shared__ float ssum[16];

  float rmax = -INFINITY, rsum = 0.0f;   // valid in lanes 0..15 (row = lane)
  v8f acc0 = {}, acc1 = {};

  // Q operand (constant over the key loop): A(16 x ETK=16) -> 4 K-groups
  v2f aq[4];
  const int qrow = qt * 16 + lm;
#pragma unroll
  for (int g = 0; g < 4; ++g)
#pragma unroll
    for (int j = 0; j < 2; ++j)
      aq[g][j] = qh[(size_t)qrow * ET_ + h * ETK_ + g * 4 + hlf * 2 + j];

  const float* kb = kh + (size_t)b * LK_ * ET_;
  const float* xb = x  + (size_t)b * LK_ * D_;

  for (int k0 = 0; k0 < LK_; k0 += 16) {
    // ---- scores tile: S = Q . K^T / 4  (16q x 16k) ----
    v8f sc = {};
#pragma unroll
    for (int g = 0; g < 4; ++g) {
      v2f bk;
#pragma unroll
      for (int j = 0; j < 2; ++j)
        bk[j] = kb[(size_t)(k0 + lm) * ET_ + h * ETK_ + g * 4 + hlf * 2 + j];
      sc = wmma_f32(aq[g], bk, sc);
    }
#pragma unroll
    for (int j = 0; j < 8; ++j)
      sS[(j + 8 * hlf) * 16 + lm] = sc[j] * 0.25f;
    __syncthreads();

    // ---- online softmax (lanes 0..15 each own one row) ----
    if (lane < 16) {
      float tmax = -INFINITY;
      for (int n = 0; n < 16; ++n) tmax = fmaxf(tmax, sS[lane * 16 + n]);
      const float nm = fmaxf(rmax, tmax);
      const float corr = __expf(rmax - nm);   // 0 on first tile (rmax=-inf)
      float tsum = 0.0f;
      for (int n = 0; n < 16; ++n) {
        const float p = __expf(sS[lane * 16 + n] - nm);
        sS[lane * 16 + n] = p;
        tsum += p;
      }
      rsum = rsum * corr + tsum;
      rmax = nm;
      scorr[lane] = corr;
    }
    __syncthreads();

    // ---- rescale accumulators by per-row correction ----
#pragma unroll
    for (int j = 0; j < 8; ++j) {
      const float f = scorr[j + 8 * hlf];
      acc0[j] *= f;
      acc1[j] *= f;
    }

    // ---- acc += P(16x16) @ X(16x32) ----
#pragma unroll
    for (int g = 0; g < 4; ++g) {
      v2f ap, bx0, bx1;
#pragma unroll
      for (int j = 0; j < 2; ++j) {
        const int kk = g * 4 + hlf * 2 + j;
        ap[j]  = sS[lm * 16 + kk];
        bx0[j] = xb[(size_t)(k0 + kk) * D_ + lm];
        bx1[j] = xb[(size_t)(k0 + kk) * D_ + 16 + lm];
      }
      acc0 = wmma_f32(ap, bx0, acc0);
      acc1 = wmma_f32(ap, bx1, acc1);
    }
    __syncthreads();   // protect sS before next tile overwrites it
  }

  if (lane < 16) ssum[lane] = rsum;
  __syncthreads();
#pragma unroll
  for (int j = 0; j < 8; ++j) {
    const float inv = 1.0f / ssum[j + 8 * hlf];
    const int q = qt * 16 + j + 8 * hlf;
    float* dst = att + ((size_t)b * LQ_ + q) * (H_ * D_) + h * D_;
    dst[lm]      = acc0[j] * inv;
    dst[16 + lm] = acc1[j] * inv;
  }
}

// ---------------------------------------------------------------------------
// GRU over LQ=128 steps, single workgroup of 8 waves. h and gate buffer live
// in LDS (64 KB total); h @ W_hh^T done with WMMA (48 16x16 tiles / step).
// ---------------------------------------------------------------------------
__global__ void gru_kernel(const float* __restrict__ xg,   // (B, LQ, 3*NH)
                           const float* __restrict__ Whh,  // (3*NH, NH)
                           const float* __restrict__ bhh,  // (3*NH)
                           float* __restrict__ hout) {     // (B, NH)
  __shared__ float h_s[B_ * NH_];        // 16 KB
  __shared__ float hg_s[B_ * 3 * NH_];   // 48 KB
  const int tid = threadIdx.x;
  const int wave = tid >> 5, lane = tid & 31, lm = lane & 15, hlf = (lane >> 4) & 1;

  for (int i = tid; i < B_ * NH_; i += 256) h_s[i] = 0.0f;
  __syncthreads();

  for (int t = 0; t < LQ_; ++t) {
    // Phase A: hg = h @ Whh^T  (M=32 -> 2 m-tiles, N=384 -> 24 n-tiles)
    for (int tile = wave; tile < 48; tile += 8) {
      const int mt = tile & 1, nt = tile >> 1;
      const int m0 = mt * 16, n0 = nt * 16;
      v8f acc = {};
      for (int g = 0; g < NH_ / 4; ++g) {
        v2f a, w;
#pragma unroll
        for (int j = 0; j < 2; ++j) {
          const int kk = g * 4 + hlf * 2 + j;
          a[j] = h_s[(m0 + lm) * NH_ + kk];
          w[j] = Whh[(size_t)(n0 + lm) * NH_ + kk];
        }
        acc = wmma_f32(a, w, acc);
      }
#pragma unroll
      for (int j = 0; j < 8; ++j)
        hg_s[(m0 + j + 8 * hlf) * (3 * NH_) + n0 + lm] = acc[j];
    }
    __syncthreads();

    // Phase B: gate nonlinearities + hidden update
    for (int i = tid; i < B_ * NH_; i += 256) {
      const int bb = i >> 7, c = i & (NH_ - 1);
      const float* xrow = xg + ((size_t)bb * LQ_ + t) * (3 * NH_);
      const float hr = hg_s[bb * (3 * NH_) + c]            + bhh[c];
      const float hz = hg_s[bb * (3 * NH_) + NH_ + c]      + bhh[NH_ + c];
      const float hn = hg_s[bb * (3 * NH_) + 2 * NH_ + c]  + bhh[2 * NH_ + c];
      const float r = 1.0f / (1.0f + __expf(-(xrow[c] + hr)));
      const float z = 1.0f / (1.0f + __expf(-(xrow[NH_ + c] + hz)));
      const float n = tanhf(xrow[2 * NH_ + c] + r * hn);
      h_s[i] = (1.0f - z) * n + z * h_s[i];
    }
    __syncthreads();
  }

  for (int i = tid; i < B_ * NH_; i += 256) hout[i] = h_s[i];
}

// ---------------------------------------------------------------------------
extern "C" void kernel_launch(void* const* d_in, const int* in_sizes, int n_in,
                              void* d_out, int out_size, void* d_ws, size_t ws_size,
                              hipStream_t stream) {
  (void)in_sizes; (void)n_in; (void)out_size; (void)ws_size;

  const float* x      = (const float*)d_in[0];   // (B, LK, D)
  const float* tsteps = (const float*)d_in[1];   // (B, LK)
  const float* qtimes = (const float*)d_in[2];   // (LQ)
  const float* Wlin   = (const float*)d_in[3];
  const float* blin   = (const float*)d_in[4];
  const float* Wper   = (const float*)d_in[5];
  const float* bper   = (const float*)d_in[6];
  const float* Wq     = (const float*)d_in[7];
  const float* bq     = (const float*)d_in[8];
  const float* Wk     = (const float*)d_in[9];
  const float* bk     = (const float*)d_in[10];
  const float* Wo     = (const float*)d_in[11];
  const float* bo     = (const float*)d_in[12];
  const float* W_ih   = (const float*)d_in[13];
  const float* b_ih   = (const float*)d_in[14];
  const float* W_hh   = (const float*)d_in[15];
  const float* b_hh   = (const float*)d_in[16];
  const float* W1     = (const float*)d_in[17];
  const float* b1     = (const float*)d_in[18];
  const float* W2     = (const float*)d_in[19];
  const float* b2     = (const float*)d_in[20];
  const float* W3     = (const float*)d_in[21];
  const float* b3     = (const float*)d_in[22];

  // workspace layout (floats)
  float* w    = (float*)d_ws;
  float* qh   = w;                               // LQ*ET            = 16384
  float* kh   = qh  + (size_t)LQ_ * ET_;         // B*LK*ET          = 8388608
  float* att  = kh  + (size_t)B_ * LK_ * ET_;    // B*LQ*H*D         = 1048576
  float* outp = att + (size_t)B_ * LQ_ * H_ * D_;// B*LQ*NH          = 524288
  float* xg   = outp+ (size_t)B_ * LQ_ * NH_;    // B*LQ*3NH         = 1572864
  float* hfin = xg  + (size_t)B_ * LQ_ * 3*NH_;  // B*NH             = 4096
  float* c1   = hfin+ (size_t)B_ * NH_;          // 32*300           = 9600
  float* c2   = c1  + (size_t)B_ * 300;          // 32*300           = 9600

  // 1) query embedding + Q projection  (M = LQ = 128)
  proj_embed_gemm<<<dim3(LQ_ / 16, ET_ / 16), 32, 0, stream>>>(
      qtimes, Wlin, blin, Wper, bper, Wq, bq, qh, LQ_);

  // 2) key embedding + K projection  (M = B*LK = 65536)
  proj_embed_gemm<<<dim3(B_ * LK_ / 16, ET_ / 16), 32, 0, stream>>>(
      tsteps, Wlin, blin, Wper, bper, Wk, bk, kh, B_ * LK_);

  // 3) attention with online softmax
  attention_kernel<<<dim3(LQ_ / 16, H_, B_), 32, 0, stream>>>(qh, kh, x, att);

  // 4) output projection: (B*LQ, 256) @ Wo^T -> (B*LQ, 128)
  gemm_wmma<<<dim3(B_ * LQ_ / 16, NH_ / 16), 32, 0, stream>>>(
      att, Wo, bo, outp, B_ * LQ_, NH_, H_ * D_, 0);

  // 5) GRU input gates: (B*LQ, 128) @ W_ih^T -> (B*LQ, 384)
  gemm_wmma<<<dim3(B_ * LQ_ / 16, 3 * NH_ / 16), 32, 0, stream>>>(
      outp, W_ih, b_ih, xg, B_ * LQ_, 3 * NH_, NH_, 0);

  // 6) GRU recurrence (single workgroup, 8 waves)
  gru_kernel<<<1, 256, 0, stream>>>(xg, W_hh, b_hh, hfin);

  // 7) classifier: 128 -> 300 (relu) -> 300 (relu) -> 2
  gemm_wmma<<<dim3(2, 19), 32, 0, stream>>>(hfin, W1, b1, c1, B_, 300, NH_, 1);
  gemm_wmma<<<dim3(2, 19), 32, 0, stream>>>(c1,   W2, b2, c2, B_, 300, 300, 1);
  gemm_wmma<<<dim3(2, 1),  32, 0, stream>>>(c2,   W3, b3, (float*)d_out, B_, 2, 300, 0);
}